// RNNWithAttention_89885075570790
// MI455X (gfx1250) — compile-verified
//
#include <hip/hip_runtime.h>
#include <hip/hip_bf16.h>
#include <math.h>

// ---------------------------------------------------------------------------
// RNN with Bahdanau attention, MI455X (gfx1250, wave32, WMMA).
// T=2048, E=512, H=2048, V=128.
// ---------------------------------------------------------------------------

#define T_SEQ 2048
#define EMB   512
#define HID   2048
#define VOC   128

typedef float v2f __attribute__((ext_vector_type(2)));
typedef float v8f __attribute__((ext_vector_type(8)));

// ---------------------------------------------------------------------------
// K-segment of a 2x2-tiled fp32 WMMA GEMM.
//   Ap0/Ap1 : per-lane A row base pointers for the two M tiles, already
//             offset so local k=0 is the segment start.
//   Bp      : per-lane B pointer, already offset to
//             (kstart + 2*half)*N + n0 + r.
//   len     : segment length in K (multiple of 4).
// Each iteration: 2 A b64 loads + 4 B b32 loads -> 4 independent WMMAs.
// ---------------------------------------------------------------------------
__device__ __forceinline__ void gemm_seg_f32(
    v8f& acc00, v8f& acc01, v8f& acc10, v8f& acc11,
    const float* __restrict__ Ap0, const float* __restrict__ Ap1,
    const float* __restrict__ Bp, int len, int N, int half)
{
    int ka = half << 1;                       // local k for this lane-half
    for (int k = 0; k < len; k += 4) {
        if ((k & 63) == 0) {
            // CDNA5 speculative prefetch of B ahead (drops silently if OOB)
            __builtin_prefetch(Bp + (size_t)64 * N, 0, 1);
        }
        v2f a0 = *(const v2f*)(Ap0 + ka);
        v2f a1 = *(const v2f*)(Ap1 + ka);
        v2f b0, b1;
        b0.x = Bp[0];
        b0.y = Bp[N];
        b1.x = Bp[16];
        b1.y = Bp[N + 16];
        acc00 = __builtin_amdgcn_wmma_f32_16x16x4_f32(false, a0, false, b0,
                                                      (short)0, acc00, false, false);
        acc01 = __builtin_amdgcn_wmma_f32_16x16x4_f32(false, a0, false, b1,
                                                      (short)0, acc01, false, false);
        acc10 = __builtin_amdgcn_wmma_f32_16x16x4_f32(false, a1, false, b0,
                                                      (short)0, acc10, false, false);
        acc11 = __builtin_amdgcn_wmma_f32_16x16x4_f32(false, a1, false, b1,
                                                      (short)0, acc11, false, false);
        ka += 4;
        Bp += (size_t)4 * N;
    }
}

__device__ __forceinline__ void store_tile_f32(
    const v8f& acc, float* __restrict__ C, int N,
    int row0, int col, const float* __restrict__ bias, int applyTanh)
{
    #pragma unroll
    for (int v = 0; v < 8; ++v) {
        float val = acc[v];
        if (bias)      val += bias[col];
        if (applyTanh) val  = tanhf(val);
        C[(size_t)(row0 + v) * (size_t)N + col] = val;
    }
}

// ---------------------------------------------------------------------------
// Generic fp32 WMMA GEMM:  C[M,N] = act( A[M,K] @ B[K,N] + bias )
//   A split: columns [0,Ksplit) from A0, [Ksplit,K) from A1
//   (pass A1=A0, Ksplit=K for one matrix).
//   rowidx (optional): A0 logical row m reads physical row rowidx[m]
//   (fuses the embedding gather).
// One wave computes a 32x32 C tile as 2x2 WMMA 16x16 tiles.
// M, N multiples of 32; K, Ksplit multiples of 4.
// ---------------------------------------------------------------------------
__global__ void __launch_bounds__(256)
rnnatt_gemm_wmma_f32(const float* __restrict__ A0,
                     const float* __restrict__ A1, int Ksplit,
                     const int*   __restrict__ rowidx,
                     const float* __restrict__ B,
                     const float* __restrict__ bias,
                     float* __restrict__ C,
                     int M, int N, int K, int applyTanh)
{
    const int lane  = threadIdx.x & 31;
    const int wave  = threadIdx.x >> 5;
    const int tilesN = N >> 5;                         // 32-wide tiles
    const int totalTiles = (M >> 5) * tilesN;
    const int tile = blockIdx.x * (blockDim.x >> 5) + wave;
    if (tile >= totalTiles) return;

    const int m0   = (tile / tilesN) << 5;
    const int n0   = (tile % tilesN) << 5;
    const int half = lane >> 4;       // 0: K {0,1}, 1: K {2,3} per k-step
    const int r    = lane & 15;

    const int row0 = m0 + r;          // M tile 0 row for this lane
    const int row1 = m0 + 16 + r;     // M tile 1 row
    const int col  = n0 + r;          // N tile 0 column

    const int arow0 = rowidx ? rowidx[row0] : row0;
    const int arow1 = rowidx ? rowidx[row1] : row1;

    v8f acc00 = {}, acc01 = {}, acc10 = {}, acc11 = {};

    // ---- segment 1: A0, k in [0, Ksplit) ----
    gemm_seg_f32(acc00, acc01, acc10, acc11,
                 A0 + (size_t)arow0 * Ksplit,
                 A0 + (size_t)arow1 * Ksplit,
                 B + (size_t)(half << 1) * N + col,
                 Ksplit, N, half);

    // ---- segment 2: A1, k in [Ksplit, K) ----
    const int len2 = K - Ksplit;
    if (len2 > 0) {
        gemm_seg_f32(acc00, acc01, acc10, acc11,
                     A1 + (size_t)row0 * len2,
                     A1 + (size_t)row1 * len2,
                     B + (size_t)(Ksplit + (half << 1)) * N + col,
                     len2, N, half);
    }

    // ---- epilogue: D layout VGPR v -> row +v+8*half, col lane&15 ----
    const int rbase = half << 3;
    store_tile_f32(acc00, C, N, m0 + rbase,      col,      bias, applyTanh);
    store_tile_f32(acc01, C, N, m0 + rbase,      col + 16, bias, applyTanh);
    store_tile_f32(acc10, C, N, m0 + 16 + rbase, col,      bias, applyTanh);
    store_tile_f32(acc11, C, N, m0 + 16 + rbase, col + 16, bias, applyTanh);
}

// ---------------------------------------------------------------------------
// Init: h_cur = h0, reset grid-barrier counter.
// ---------------------------------------------------------------------------
__global__ void rnnatt_init(const float* __restrict__ h0,
                            float* __restrict__ h_cur,
                            unsigned* __restrict__ cnt)
{
    const int g = blockIdx.x * blockDim.x + threadIdx.x;
    if (g < HID) h_cur[g] = h0[g];
    if (g == 0)  *cnt = 0u;
}

// ---------------------------------------------------------------------------
// Grid-wide barrier via monotonically increasing atomic counter.
// ---------------------------------------------------------------------------
__device__ __forceinline__ void rnnatt_grid_barrier(unsigned* cnt, unsigned target)
{
    __syncthreads();
    if (threadIdx.x == 0) {
        __threadfence();
        atomicAdd(cnt, 1u);
        while (__hip_atomic_load(cnt, __ATOMIC_RELAXED,
                                 __HIP_MEMORY_SCOPE_AGENT) < target) { }
        __threadfence();
    }
    __syncthreads();
}

#define SCAN_NB 32

// ---------------------------------------------------------------------------
// Persistent recurrent scan.  32 blocks x 256 threads = 8192 lanes.
// Phase 1: 4 K-chunks x 2048 outputs of the GEMV h @ W_hh (coalesced rows,
//          W_hh stays L2-resident: 16 MB << 192 MB).
// Phase 2: reduce chunks, add xh_t + b, tanh, store H[t] / carry h.
// ---------------------------------------------------------------------------
__global__ void __launch_bounds__(256)
rnnatt_scan(const float* __restrict__ xh,     // (T,H)
            const float* __restrict__ W_hh,   // (H,H)
            const float* __restrict__ b_hh,   // (H)
            float* __restrict__ Hout,         // (T,H)
            float* __restrict__ h_cur,        // (H)
            float* __restrict__ partial,      // (4,H)
            unsigned* __restrict__ cnt)
{
    const unsigned NB = gridDim.x;
    const int g     = blockIdx.x * 256 + threadIdx.x;   // 0..8191
    const int chunk = g >> 11;                          // i-chunk 0..3
    const int j     = g & (HID - 1);                    // output column
    const int i0    = chunk << 9;                       // 512 i's per chunk
    unsigned bar = 0;

    for (int t = 0; t < T_SEQ; ++t) {
        float s = 0.f;
        const float* __restrict__ col = W_hh + (size_t)i0 * HID + j;
        #pragma unroll 4
        for (int i = 0; i < 512; ++i) {
            s = fmaf(h_cur[i0 + i], col[(size_t)i * HID], s);
        }
        partial[(chunk << 11) + j] = s;

        ++bar; rnnatt_grid_barrier(cnt, NB * bar);

        if (g < HID) {
            float tot = partial[g] + partial[HID + g]
                      + partial[2 * HID + g] + partial[3 * HID + g]
                      + xh[(size_t)t * HID + g] + b_hh[g];
            float hn = tanhf(tot);
            h_cur[g] = hn;
            Hout[(size_t)t * HID + g] = hn;
        }

        ++bar; rnnatt_grid_barrier(cnt, NB * bar);
    }
}

// ---------------------------------------------------------------------------
// scores[t] = dot(Aatt[t,:], v_a)  — one block per row, deterministic tree.
// ---------------------------------------------------------------------------
__global__ void __launch_bounds__(256)
rnnatt_scores(const float* __restrict__ Aatt,  // (T,H) = tanh(H @ W_a)
              const float* __restrict__ v_a,   // (H)
              float* __restrict__ scores)      // (T)
{
    __shared__ float sm[256];
    const int t   = blockIdx.x;
    const int tid = threadIdx.x;
    const float* __restrict__ row = Aatt + (size_t)t * HID;
    float s = 0.f;
    for (int c = tid; c < HID; c += 256) s = fmaf(row[c], v_a[c], s);
    sm[tid] = s;
    __syncthreads();
    for (int off = 128; off > 0; off >>= 1) {
        if (tid < off) sm[tid] += sm[tid + off];
        __syncthreads();
    }
    if (tid == 0) scores[t] = sm[0];
}

// ---------------------------------------------------------------------------
// Softmax prep: global max, e[s] = exp(scores[s]-M), invZ[t] = 1/cumsum(e).
// Single block; serial prefix (T=2048, trivial) for determinism.
// ---------------------------------------------------------------------------
__global__ void __launch_bounds__(256)
rnnatt_softmax_prep(const float* __restrict__ scores,
                    float* __restrict__ e,
                    float* __restrict__ invZ)
{
    __shared__ float sm[256];
    const int tid = threadIdx.x;
    float m = -3.4e38f;
    for (int s = tid; s < T_SEQ; s += 256) m = fmaxf(m, scores[s]);
    sm[tid] = m;
    __syncthreads();
    for (int off = 128; off > 0; off >>= 1) {
        if (tid < off) sm[tid] = fmaxf(sm[tid], sm[tid + off]);
        __syncthreads();
    }
    const float M = sm[0];
    __syncthreads();
    for (int s = tid; s < T_SEQ; s += 256) e[s] = expf(scores[s] - M);
    __syncthreads();
    if (tid == 0) {
        float run = 0.f;
        for (int t = 0; t < T_SEQ; ++t) {
            run += e[t];
            invZ[t] = 1.0f / run;
        }
    }
}

// ---------------------------------------------------------------------------
// Causal attention as a prefix scan (scores independent of query):
//   context[t,c] = invZ[t] * sum_{s<=t} e[s] * H[s,c]
// 2048 lanes, one column each; fully coalesced row-sweeps.
// ---------------------------------------------------------------------------
__global__ void __launch_bounds__(256)
rnnatt_context(const float* __restrict__ Hbuf,  // (T,H)
               const float* __restrict__ e,     // (T)
               const float* __restrict__ invZ,  // (T)
               float* __restrict__ ctx)         // (T,H)
{
    const int c = blockIdx.x * blockDim.x + threadIdx.x;  // column
    float acc = 0.f;
    for (int t = 0; t < T_SEQ; ++t) {
        acc = fmaf(e[t], Hbuf[(size_t)t * HID + c], acc);
        ctx[(size_t)t * HID + c] = acc * invZ[t];
    }
}

// ---------------------------------------------------------------------------
// h_last = H[T-1, :]
// ---------------------------------------------------------------------------
__global__ void rnnatt_hlast(const float* __restrict__ Hbuf,
                             float* __restrict__ out)
{
    const int g = blockIdx.x * blockDim.x + threadIdx.x;
    if (g < HID) out[g] = Hbuf[(size_t)(T_SEQ - 1) * HID + g];
}

// ---------------------------------------------------------------------------
// Launcher.  Workspace layout (floats):
//   buf0 : xh, then reused for Aatt = tanh(H @ W_a)    (T*H)
//   buf1 : H                                           (T*H)
//   buf2 : context                                     (T*H)
//   scores / e / invZ / h_cur (2048 each), partial (4*2048), counter
// ---------------------------------------------------------------------------
extern "C" void kernel_launch(void* const* d_in, const int* in_sizes, int n_in,
                              void* d_out, int out_size, void* d_ws, size_t ws_size,
                              hipStream_t stream)
{
    const int*   x     = (const int*)  d_in[0];
    const float* h0    = (const float*)d_in[1];
    const float* emb   = (const float*)d_in[2];
    const float* W_xh  = (const float*)d_in[3];
    const float* W_hh  = (const float*)d_in[4];
    const float* b_hh  = (const float*)d_in[5];
    const float* W_a   = (const float*)d_in[6];
    const float* v_a   = (const float*)d_in[7];
    const float* W_out = (const float*)d_in[8];
    const float* b_out = (const float*)d_in[9];

    float* out = (float*)d_out;                 // logits (T*V) then h_last (H)
    float* ws  = (float*)d_ws;

    const size_t TH = (size_t)T_SEQ * HID;      // 4,194,304
    float* buf0   = ws;                         // xh, later Aatt
    float* buf1   = buf0 + TH;                  // H
    float* buf2   = buf1 + TH;                  // context
    float* scores = buf2 + TH;
    float* evec   = scores + T_SEQ;
    float* invZ   = evec + T_SEQ;
    float* h_cur  = invZ + T_SEQ;
    float* partial = h_cur + HID;               // 4*HID
    unsigned* cnt = (unsigned*)(partial + 4 * HID);

    // 0) init carry + barrier counter
    hipLaunchKernelGGL(rnnatt_init, dim3(8), dim3(256), 0, stream,
                       h0, h_cur, cnt);

    // 1) xh = emb[x] @ W_xh        M=2048 N=2048 K=512  (gather fused)
    {
        int tiles = (T_SEQ / 32) * (HID / 32);  // 4096 wave-tiles
        hipLaunchKernelGGL(rnnatt_gemm_wmma_f32, dim3(tiles / 8), dim3(256), 0, stream,
                           emb, emb, EMB, x, W_xh, (const float*)nullptr,
                           buf0, T_SEQ, HID, EMB, 0);
    }

    // 2) recurrent scan -> H (buf1)
    hipLaunchKernelGGL(rnnatt_scan, dim3(SCAN_NB), dim3(256), 0, stream,
                       buf0, W_hh, b_hh, buf1, h_cur, partial, cnt);

    // 3) Aatt = tanh(H @ W_a)      M=2048 N=2048 K=2048   (reuses buf0)
    {
        int tiles = (T_SEQ / 32) * (HID / 32);
        hipLaunchKernelGGL(rnnatt_gemm_wmma_f32, dim3(tiles / 8), dim3(256), 0, stream,
                           buf1, buf1, HID, (const int*)nullptr, W_a,
                           (const float*)nullptr, buf0, T_SEQ, HID, HID, 1);
    }

    // 4) scores = Aatt @ v_a
    hipLaunchKernelGGL(rnnatt_scores, dim3(T_SEQ), dim3(256), 0, stream,
                       buf0, v_a, scores);

    // 5) softmax prep (max / exp / prefix Z)
    hipLaunchKernelGGL(rnnatt_softmax_prep, dim3(1), dim3(256), 0, stream,
                       scores, evec, invZ);

    // 6) causal context via prefix scan
    hipLaunchKernelGGL(rnnatt_context, dim3(HID / 256), dim3(256), 0, stream,
                       buf1, evec, invZ, buf2);

    // 7) logits = [H | context] @ W_out + b_out   M=2048 N=128 K=4096
    {
        int tiles = (T_SEQ / 32) * (VOC / 32);  // 256 wave-tiles
        hipLaunchKernelGGL(rnnatt_gemm_wmma_f32, dim3(tiles / 8), dim3(256), 0, stream,
                           buf1, buf2, HID, (const int*)nullptr, W_out,
                           b_out, out, T_SEQ, VOC, 2 * HID, 0);
    }

    // 8) h_last
    hipLaunchKernelGGL(rnnatt_hlast, dim3(8), dim3(256), 0, stream,
                       buf1, out + (size_t)T_SEQ * VOC);
}